// ConstraintProcessor_61246233641261
// MI455X (gfx1250) — compile-verified
//
#include <hip/hip_runtime.h>

// ---------------------------------------------------------------------------
// GCN (3 layers + output projection), d = 64, fp32 throughout.
// GEMMs use CDNA5 fp32 WMMA: v_wmma_f32_16x16x4_f32.
// Edge aggregation uses L2-resident f32 global atomics (features = 25.6 MB
// fits in MI455X's 192 MB L2).
// ---------------------------------------------------------------------------

typedef __attribute__((ext_vector_type(2))) float v2f;
typedef __attribute__((ext_vector_type(8))) float v8f;

#define D 64

// --- graph normalization --------------------------------------------------

__global__ void init_deg(float* __restrict__ deg, int N) {
  int i = blockIdx.x * blockDim.x + threadIdx.x;
  if (i < N) deg[i] = 1.0f;  // self-loop contributes 1 to every node's degree
}

__global__ void accum_deg(const long long* __restrict__ dst,
                          float* __restrict__ deg, int E) {
  int e = blockIdx.x * blockDim.x + threadIdx.x;
  if (e < E) unsafeAtomicAdd(&deg[(int)dst[e]], 1.0f);
}

__global__ void make_dinv(float* __restrict__ deg_dinv, int N) {
  int i = blockIdx.x * blockDim.x + threadIdx.x;
  if (i < N) {
    float d = deg_dinv[i];
    deg_dinv[i] = (d > 0.0f) ? rsqrtf(d) : 0.0f;
  }
}

__global__ void make_norm(const long long* __restrict__ src,
                          const long long* __restrict__ dst,
                          const float* __restrict__ dinv,
                          float* __restrict__ norm, int E) {
  int e = blockIdx.x * blockDim.x + threadIdx.x;
  if (e < E) norm[e] = dinv[(int)src[e]] * dinv[(int)dst[e]];
}

// --- fp32 WMMA GEMM: Out[N,64] = A[N,64] @ W[64,64] (+ bias) ---------------
// One wave per 16x16 output tile; 8 waves (256 thr) = 32 rows x 64 cols slab.
// A-frag layout (16x4 f32): lanes 0-15 -> M=lane, K = k0+{0,1};
//                           lanes 16-31 -> M=lane-16, K = k0+{2,3}.
// B-frag layout (4x16 f32): mirrored with N = lane&15.
// C/D layout: VGPR j, lanes 0-15 -> M=j, N=lane; lanes 16-31 -> M=j+8.

__global__ __launch_bounds__(256)
void gemm_d64_wmma(const float* __restrict__ A, const float* __restrict__ W,
                   const float* __restrict__ bias, float* __restrict__ Out,
                   int N) {
  const int wave = threadIdx.x >> 5;
  const int lane = threadIdx.x & 31;
  const int rowTile = blockIdx.x * 2 + (wave >> 2);
  const int m0 = rowTile * 16;
  if (m0 >= N) return;  // wave-uniform: EXEC stays all-ones for WMMA
  const int n0 = (wave & 3) * 16;
  const int lm = lane & 15;  // M index for A-frag, N index for B/C/D-frags
  const int kh = lane >> 4;  // K-half selector

  v8f acc = {0.f, 0.f, 0.f, 0.f, 0.f, 0.f, 0.f, 0.f};
  const float* arow = A + (size_t)(m0 + lm) * D;

#pragma unroll
  for (int k0 = 0; k0 < D; k0 += 4) {
    const int ka = k0 + kh * 2;
    v2f a, b;
    a.x = arow[ka];
    a.y = arow[ka + 1];
    b.x = W[(size_t)ka * D + n0 + lm];
    b.y = W[(size_t)(ka + 1) * D + n0 + lm];
    acc = __builtin_amdgcn_wmma_f32_16x16x4_f32(
        /*neg_a=*/false, a, /*neg_b=*/false, b,
        /*c_mod=*/(short)0, acc, /*reuse_a=*/false, /*reuse_b=*/false);
  }

  const int nn = n0 + lm;
  const float bv = bias ? bias[nn] : 0.0f;
#pragma unroll
  for (int j = 0; j < 8; ++j) {
    const int m = m0 + kh * 8 + j;
    Out[(size_t)m * D + nn] = acc[j] + bv;
  }
}

// --- aggregation ----------------------------------------------------------

// agg[i] = dinv[i]^2 * h[i]   (self-loop term; also zero-initializes agg)
__global__ void self_loop_init(const float* __restrict__ h,
                               const float* __restrict__ dinv,
                               float* __restrict__ agg, int total) {
  int t = blockIdx.x * blockDim.x + threadIdx.x;
  if (t < total) {
    float di = dinv[t >> 6];
    agg[t] = di * di * h[t];
  }
}

// agg[dst] += norm * h[src] ; 16 lanes per edge, float4 per lane.
__global__ void scatter_edges(const long long* __restrict__ src,
                              const long long* __restrict__ dst,
                              const float* __restrict__ norm,
                              const float* __restrict__ h,
                              float* __restrict__ agg, int E) {
  long long t = (long long)blockIdx.x * blockDim.x + threadIdx.x;
  if (t >= (long long)E * 16) return;
  const int e = (int)(t >> 4);
  const int g = (int)(t & 15);
  const int s = (int)src[e];
  const int d = (int)dst[e];
  const float nrm = norm[e];
  const float4 hv = *(const float4*)(h + (size_t)s * D + g * 4);
  float* ap = agg + (size_t)d * D + g * 4;
  unsafeAtomicAdd(ap + 0, nrm * hv.x);
  unsafeAtomicAdd(ap + 1, nrm * hv.y);
  unsafeAtomicAdd(ap + 2, nrm * hv.z);
  unsafeAtomicAdd(ap + 3, nrm * hv.w);
}

__global__ void bias_relu(const float* __restrict__ agg,
                          const float* __restrict__ b,
                          float* __restrict__ xout, int total) {
  int t = blockIdx.x * blockDim.x + threadIdx.x;
  if (t < total) {
    float v = agg[t] + b[t & (D - 1)];
    xout[t] = v > 0.0f ? v : 0.0f;
  }
}

// --- driver ---------------------------------------------------------------

extern "C" void kernel_launch(void* const* d_in, const int* in_sizes, int n_in,
                              void* d_out, int out_size, void* d_ws,
                              size_t ws_size, hipStream_t stream) {
  const float* x         = (const float*)d_in[0];
  const long long* edge  = (const long long*)d_in[1];  // int64 in reference
  const float* Ws        = (const float*)d_in[2];
  const float* bs        = (const float*)d_in[3];
  const float* W_out     = (const float*)d_in[4];
  const float* b_out     = (const float*)d_in[5];

  const int N = in_sizes[0] / D;
  const int E = in_sizes[1] / 2;
  const long long* srcv = edge;      // edge_index[0]
  const long long* dstv = edge + E;  // edge_index[1]

  // scratch layout (all 16B-aligned given N,E multiples of 4)
  char* ws = (char*)d_ws;
  float* dinv = (float*)ws; ws += sizeof(float) * (size_t)N;
  float* norm = (float*)ws; ws += sizeof(float) * (size_t)E;
  float* h    = (float*)ws; ws += sizeof(float) * (size_t)N * D;
  float* agg  = (float*)ws; ws += sizeof(float) * (size_t)N * D;
  float* xbuf = (float*)ws;

  const int T = 256;
  const int total = N * D;
  const int rowTiles = (N + 15) / 16;
  const int gemmBlocks = (rowTiles + 1) / 2;

  // graph normalization (deterministic, recomputed every call)
  init_deg<<<(N + T - 1) / T, T, 0, stream>>>(dinv, N);
  accum_deg<<<(E + T - 1) / T, T, 0, stream>>>(dstv, dinv, E);
  make_dinv<<<(N + T - 1) / T, T, 0, stream>>>(dinv, N);
  make_norm<<<(E + T - 1) / T, T, 0, stream>>>(srcv, dstv, dinv, norm, E);

  const long long st = (long long)E * 16;
  const unsigned scatterBlocks = (unsigned)((st + T - 1) / T);

  const float* xin = x;
  for (int i = 0; i < 3; ++i) {
    gemm_d64_wmma<<<gemmBlocks, 256, 0, stream>>>(
        xin, Ws + (size_t)i * D * D, nullptr, h, N);
    self_loop_init<<<(total + T - 1) / T, T, 0, stream>>>(h, dinv, agg, total);
    scatter_edges<<<scatterBlocks, T, 0, stream>>>(srcv, dstv, norm, h, agg, E);
    bias_relu<<<(total + T - 1) / T, T, 0, stream>>>(agg, bs + (size_t)i * D,
                                                     xbuf, total);
    xin = xbuf;
  }
  gemm_d64_wmma<<<gemmBlocks, 256, 0, stream>>>(xin, W_out, b_out,
                                                (float*)d_out, N);
}